// SVDNN_22428319220110
// MI455X (gfx1250) — compile-verified
//
#include <hip/hip_runtime.h>
#include <hip/hip_bf16.h>
#include <math.h>

// ---------------------------------------------------------------------------
// MI455X (gfx1250) implementation.
//
// Roofline-driven design:
//  * h [N,1344] never materialized (537MB): h = [x | diag(c_j) B_i] with 5
//    SpMM outputs B0..B4 [N,64] + 4 scale vectors. Working set ~180MB ->
//    L2-resident (192MB), dense phase streams from L2 not HBM.
//  * Tall-skinny QR via CholeskyQR (WMMA Gram + tiny chol/tri-inv kernel).
//  * [60,1344] SVD via 60x60 Jacobi of Zf^T Zf; s cancels:
//    term = Q @ (ub50 ub50^T (Zf^T w)).
//  * fp32 WMMA (V_WMMA_F32_16X16X4_F32) for all tall GEMMs.
//  * Row tiles for WMMA A-operands staged in LDS via the Tensor Data Mover
//    (tensor_load_to_lds, TENSORcnt waits, LDS pad for bank-conflict-free
//    pitch-65 rows), double-buffered in the hot h@V kernel so the next
//    source matrix DMAs in while WMMA consumes the current one.
// ---------------------------------------------------------------------------

typedef __attribute__((ext_vector_type(2))) float v2f;
typedef __attribute__((ext_vector_type(8))) float v8f;

#define KD    60      // rank + oversampling
#define RANK  50
#define HIDN  32
#define NCLS  696
#define ALPHA_PR 0.85f
#define NPOW  4
#define TPITCH 65     // LDS tile pitch in floats (64 + 1 pad dword per row)

__device__ __forceinline__ v8f wmma4(v2f a, v2f b, v8f c) {
#if defined(__AMDGCN__) && __has_builtin(__builtin_amdgcn_wmma_f32_16x16x4_f32)
  return __builtin_amdgcn_wmma_f32_16x16x4_f32(
      false, a, false, b, (short)0, c, false, false);
#else
  return c;  // host pass only
#endif
}

#if defined(__AMDGCN__) && __has_builtin(__builtin_amdgcn_tensor_load_to_lds) && \
    __has_builtin(__builtin_amdgcn_s_wait_tensorcnt)
#define USE_TDM 1
#else
#define USE_TDM 0
#endif

__device__ __forceinline__ unsigned lds_off(const void* p) {
  // LDS aperture: low 32 bits of a generic pointer are the LDS byte address.
  return (unsigned)(unsigned long long)p;
}

#if USE_TDM
// DMA a [tile_rows x 64] f32 tile (row stride 64) from gptr into LDS with one
// pad dword per 64 dwords (-> pitch TPITCH). Rows >= rows_remaining read as 0.
// D# layout per CDNA5 ISA (group0/group1; 2D tile).
__device__ __forceinline__ void tdm_load_tile_f32(unsigned lds_byte,
                                                  const float* gptr,
                                                  int rows_remaining,
                                                  int tile_rows) {
  typedef unsigned int v4u __attribute__((ext_vector_type(4)));
  typedef int v8i __attribute__((ext_vector_type(8)));
  typedef int v4i __attribute__((ext_vector_type(4)));
  unsigned long long ga = (unsigned long long)gptr;
  v4u g0;
  g0.x = 1u;                                        // count=1 valid descriptor
  g0.y = lds_byte;                                  // lds_addr [63:32]
  g0.z = (unsigned)(ga & 0xFFFFFFFFu);              // global_addr [95:64]
  g0.w = (unsigned)((ga >> 32) & 0x01FFFFFFu) | (2u << 30);  // addr hi | type=2
  unsigned td1 = (unsigned)(rows_remaining > 0 ? rows_remaining : 1);
  v8i g1;
  g1[0] = (int)((2u << 16) | (1u << 20) | (5u << 22));  // data_size=4B, pad_en,
                                                        // pad every 64 dw, +1 dw
  g1[1] = (int)(64u << 16);                         // tensor_dim0=64 (lo16 here)
  g1[2] = (int)((td1 & 0xFFFFu) << 16);             // dim0 hi16=0 | tensor_dim1 lo
  g1[3] = (int)(((td1 >> 16) & 0xFFFFu) | (64u << 16));  // dim1 hi | tile_dim0=64
  g1[4] = (int)(tile_rows & 0xFFFF);                // tile_dim1 (tile_dim2=0)
  g1[5] = 64;                                       // tensor_dim0_stride = 64
  g1[6] = 0; g1[7] = 0;
  v4i z4 = {};
#if __clang_major__ >= 23
  v8i z8 = {};
  __builtin_amdgcn_tensor_load_to_lds(g0, g1, z4, z4, z8, 0);
#else
  __builtin_amdgcn_tensor_load_to_lds(g0, g1, z4, z4, 0);
#endif
}
#endif

// Cooperative fallback / non-TDM tile load (coalesced, zero-fill OOB rows).
__device__ __forceinline__ void coop_load_tile(float* dst, const float* S,
                                               int base, int tile_rows, int n,
                                               int tid, int nthr) {
  for (int f = tid; f < tile_rows * 64; f += nthr) {
    int r = f >> 6, c = f & 63;
    dst[r * TPITCH + c] = (base + r < n) ? S[(size_t)(base + r) * 64 + c] : 0.0f;
  }
}

// ---------------------------------------------------------------------------
// Graph phase
// ---------------------------------------------------------------------------
__global__ void deg_kernel(const int* __restrict__ rows, float* __restrict__ deg, int E) {
  int e = blockIdx.x * blockDim.x + threadIdx.x;
  if (e < E) atomicAdd(&deg[rows[e]], 1.0f);
}

__global__ void scales_kernel(const float* __restrict__ deg, float* __restrict__ cs, int n) {
  int i = blockIdx.x * blockDim.x + threadIdx.x;
  if (i >= n) return;
  float d = deg[i];
  float ct = 1.0f / ((d > 0.0f) ? d : 1.0f);
  cs[i] = 1.0f;
  cs[(size_t)n + i] = ct;
  cs[2 * (size_t)n + i] = ALPHA_PR * ct + (1.0f - ALPHA_PR) / (float)n;
  cs[3 * (size_t)n + i] = 1.0f / (d + 1e-5f);
}

// Y[rows[e],:] += sc[cols[e]] * X[cols[e],:]  (Y pre-zeroed); 1 thread/(edge,feat).
__global__ void spmm_kernel(const float* __restrict__ X, const float* __restrict__ sc,
                            const int* __restrict__ rows, const int* __restrict__ cols,
                            float* __restrict__ Y, int E) {
  long gt = (long)blockIdx.x * blockDim.x + threadIdx.x;
  int e = (int)(gt >> 6);
  if (e >= E) return;
  int d = (int)(gt & 63);
  int r = rows[e], c = cols[e];
  atomicAdd(&Y[(size_t)r * 64 + d], sc[c] * X[(size_t)c * 64 + d]);
}

// Deterministic pseudo-normal omega (JAX threefry not reproduced).
__device__ __forceinline__ float hash_u01(unsigned x) {
  x ^= x >> 17; x *= 0xed5ad4bbu; x ^= x >> 11; x *= 0xac4c1b51u;
  x ^= x >> 15; x *= 0x31848babu; x ^= x >> 14;
  return (float)(x >> 8) * (1.0f / 16777216.0f);
}
__global__ void omega_kernel(float* __restrict__ V, int FD) {
  int idx = blockIdx.x * blockDim.x + threadIdx.x;
  if (idx >= FD * 64) return;
  int col = idx & 63;
  float v = 0.0f;
  if (col < KD) {
    unsigned u = (unsigned)idx * 4u;
    float s = hash_u01(u + 1u) + hash_u01(u + 2u) + hash_u01(u + 3u) + hash_u01(u + 4u);
    v = (s - 2.0f) * 1.7320508f;
  }
  V[idx] = v;
}

// ---------------------------------------------------------------------------
// Y[N,64] = h @ V.  8 waves x (16 rows x 64 cols) per workgroup; source row
// tiles TDM-DMA'd into LDS (double-buffered), scales staged in LDS.
// ---------------------------------------------------------------------------
__global__ void h_times_V_kernel(const float* __restrict__ x, const float* __restrict__ Bm,
                                 const float* __restrict__ cs, const float* __restrict__ V,
                                 float* __restrict__ Y, int n) {
  __shared__ float sA[2][128 * TPITCH];
  __shared__ float sS[4][128];
  const int tid = threadIdx.x;
  const int wave = tid >> 5, lane = tid & 31;
  const int half = lane >> 4, lid = lane & 15;
  const int ka = half ? 2 : 0;
  const int base = blockIdx.x * 128;
  const int m0 = base + wave * 16;
  const int lrow = wave * 16 + lid;

  for (int f = tid; f < 4 * 128; f += 256) {      // stage scales (cs[0] = ones)
    int j = f >> 7, r = f & 127;
    sS[j][r] = (base + r < n) ? cs[(size_t)j * n + base + r] : 0.0f;
  }

  v8f acc[4] = {};
#if USE_TDM
  if (wave == 0)
    tdm_load_tile_f32(lds_off(&sA[0][0]), x + (size_t)base * 64, n - base, 128);
#endif
  for (int s = 0; s < 6; ++s) {
    const int buf = s & 1;
#if USE_TDM
    if (wave == 0) {
      if (s + 1 < 6) {   // prefetch next source while computing this one
        const float* Snxt = Bm + (size_t)s * n * 64;
        tdm_load_tile_f32(lds_off(&sA[buf ^ 1][0]), Snxt + (size_t)base * 64,
                          n - base, 128);
        __builtin_amdgcn_s_wait_tensorcnt(1);   // current tile landed
      } else {
        __builtin_amdgcn_s_wait_tensorcnt(0);
      }
    }
    __syncthreads();
#else
    {
      const float* S = (s == 0) ? x : (Bm + (size_t)(s - 1) * n * 64);
      coop_load_tile(sA[buf], S, base, 128, n, tid, 256);
    }
    __syncthreads();
#endif
    const float* As = &sA[buf][lrow * TPITCH];
    const int nj = (s == 0) ? 1 : 4;
    for (int j = 0; j < nj; ++j) {
      const float scv = sS[j][lrow];
      const int vb = (s == 0) ? 0 : (1 + (s - 1) * 4 + j);
      const float* Vb = V + (size_t)vb * 4096;
      for (int kk = 0; kk < 16; ++kk) {
        const int k0 = kk * 4;
        v2f a; a.x = As[k0 + ka] * scv;
               a.y = As[k0 + ka + 1] * scv;
#pragma unroll
        for (int nt = 0; nt < 4; ++nt) {
          v2f b; b.x = Vb[(k0 + ka) * 64 + nt * 16 + lid];
                 b.y = Vb[(k0 + ka + 1) * 64 + nt * 16 + lid];
          acc[nt] = wmma4(a, b, acc[nt]);
        }
      }
    }
    __syncthreads();   // before tile buffer is overwritten
  }
#pragma unroll
  for (int nt = 0; nt < 4; ++nt)
#pragma unroll
    for (int v = 0; v < 8; ++v) {
      int row = m0 + v + 8 * half;
      if (row < n) Y[(size_t)row * 64 + nt * 16 + lid] = acc[nt][v];
    }
}

// ---------------------------------------------------------------------------
// Z[21*64,64] += h^T @ Q : block vb = B_i^T (c_j o Q). grid=(n-chunks, 21);
// split-K reduction via atomics. Loads already coalesced along lid.
// ---------------------------------------------------------------------------
__global__ void hT_times_Q_kernel(const float* __restrict__ x, const float* __restrict__ Bm,
                                  const float* __restrict__ cs, const float* __restrict__ Q,
                                  float* __restrict__ Z, int n) {
  const int vb = blockIdx.y;
  const float* S; const float* sc;
  if (vb == 0) { S = x; sc = cs; }
  else { int s = (vb - 1) >> 2, j = (vb - 1) & 3; S = Bm + (size_t)s * n * 64; sc = cs + (size_t)j * n; }
  const int wave = threadIdx.x >> 5, lane = threadIdx.x & 31;
  const int half = lane >> 4, lid = lane & 15;
  const int ka = half ? 2 : 0;
  const int mt = wave & 3;
  const int ntb = (wave >> 2) * 2;
  v8f acc[2] = {};
  const int base = blockIdx.x * 2048;
  const int lim = (n - base < 2048) ? (n - base) : 2048;
  for (int k = 0; k < lim; k += 4) {
    int r0 = base + k + ka, r1 = r0 + 1;
    float s0 = (r0 < n) ? sc[r0] : 0.0f; if (r0 >= n) r0 = n - 1;
    float s1 = (r1 < n) ? sc[r1] : 0.0f; if (r1 >= n) r1 = n - 1;
    v2f a; a.x = S[(size_t)r0 * 64 + mt * 16 + lid];
           a.y = S[(size_t)r1 * 64 + mt * 16 + lid];
#pragma unroll
    for (int t = 0; t < 2; ++t) {
      int nc = (ntb + t) * 16 + lid;
      v2f b; b.x = Q[(size_t)r0 * 64 + nc] * s0;
             b.y = Q[(size_t)r1 * 64 + nc] * s1;
      acc[t] = wmma4(a, b, acc[t]);
    }
  }
  float* Zb = Z + (size_t)vb * 4096;
#pragma unroll
  for (int t = 0; t < 2; ++t)
#pragma unroll
    for (int v = 0; v < 8; ++v) {
      int rr = mt * 16 + v + 8 * half;
      atomicAdd(&Zb[rr * 64 + (ntb + t) * 16 + lid], acc[t][v]);
    }
}

// C[64,64] += A^T @ B over n rows (split-K with atomics).
__global__ void atb_kernel(const float* __restrict__ A, const float* __restrict__ B,
                           float* __restrict__ C, int n) {
  const int wave = threadIdx.x >> 5, lane = threadIdx.x & 31;
  const int half = lane >> 4, lid = lane & 15;
  const int ka = half ? 2 : 0;
  const int mt = wave & 3;
  const int ntb = (wave >> 2) * 2;
  v8f acc[2] = {};
  const int base = blockIdx.x * 2048;
  const int lim = (n - base < 2048) ? (n - base) : 2048;
  for (int k = 0; k < lim; k += 4) {
    int r0 = base + k + ka, r1 = r0 + 1;
    float s0 = (r0 < n) ? 1.0f : 0.0f; if (r0 >= n) r0 = n - 1;
    float s1 = (r1 < n) ? 1.0f : 0.0f; if (r1 >= n) r1 = n - 1;
    v2f a; a.x = A[(size_t)r0 * 64 + mt * 16 + lid];
           a.y = A[(size_t)r1 * 64 + mt * 16 + lid];
#pragma unroll
    for (int t = 0; t < 2; ++t) {
      int nc = (ntb + t) * 16 + lid;
      v2f b; b.x = B[(size_t)r0 * 64 + nc] * s0;
             b.y = B[(size_t)r1 * 64 + nc] * s1;
      acc[t] = wmma4(a, b, acc[t]);
    }
  }
#pragma unroll
  for (int t = 0; t < 2; ++t)
#pragma unroll
    for (int v = 0; v < 8; ++v) {
      int rr = mt * 16 + v + 8 * half;
      atomicAdd(&C[rr * 64 + (ntb + t) * 16 + lid], acc[t][v]);
    }
}

// C[n,64] = A[n,64] @ B64[64,64], A tile staged in LDS via TDM.
__global__ void gemm_n64_kernel(const float* __restrict__ A, const float* __restrict__ B64,
                                float* __restrict__ C, int n) {
  __shared__ float sA[128 * TPITCH];
  const int tid = threadIdx.x;
  const int wave = tid >> 5, lane = tid & 31;
  const int half = lane >> 4, lid = lane & 15;
  const int ka = half ? 2 : 0;
  const int base = blockIdx.x * 128;
  const int m0 = base + wave * 16;
#if USE_TDM
  if (wave == 0) {
    tdm_load_tile_f32(lds_off(&sA[0]), A + (size_t)base * 64, n - base, 128);
    __builtin_amdgcn_s_wait_tensorcnt(0);
  }
  __syncthreads();
#else
  coop_load_tile(sA, A, base, 128, n, tid, 256);
  __syncthreads();
#endif
  const float* As = &sA[(wave * 16 + lid) * TPITCH];
  v8f acc[4] = {};
  for (int kk = 0; kk < 16; ++kk) {
    const int k0 = kk * 4;
    v2f a; a.x = As[k0 + ka]; a.y = As[k0 + ka + 1];
#pragma unroll
    for (int nt = 0; nt < 4; ++nt) {
      v2f b; b.x = B64[(k0 + ka) * 64 + nt * 16 + lid];
             b.y = B64[(k0 + ka + 1) * 64 + nt * 16 + lid];
      acc[nt] = wmma4(a, b, acc[nt]);
    }
  }
#pragma unroll
  for (int nt = 0; nt < 4; ++nt)
#pragma unroll
    for (int v = 0; v < 8; ++v) {
      int row = m0 + v + 8 * half;
      if (row < n) C[(size_t)row * 64 + nt * 16 + lid] = acc[nt][v];
    }
}

// ---------------------------------------------------------------------------
// Cholesky (KDxKD) + upper-triangular inverse; Rinv diag zero in cols >= KD.
// ---------------------------------------------------------------------------
__global__ void cholinv_kernel(const float* __restrict__ G, float* __restrict__ Rinv) {
  __shared__ float R[64][64];
  const int t = threadIdx.x;  // 64 threads
  for (int i = 0; i < 64; ++i) R[i][t] = G[i * 64 + t];
  __syncthreads();
  for (int k = 0; k < KD; ++k) {
    float d = sqrtf(fmaxf(R[k][k], 1e-20f));
    __syncthreads();
    if (t >= k && t < KD) R[k][t] /= d;
    __syncthreads();
    if (t > k && t < KD) {
      float rkt = R[k][t];
      for (int i = k + 1; i <= t; ++i) R[i][t] -= R[k][i] * rkt;
    }
    __syncthreads();
  }
  float xc[64];
  if (t < KD) {
    for (int i = 0; i < 64; ++i) xc[i] = 0.0f;
    xc[t] = 1.0f / R[t][t];
    for (int i = t - 1; i >= 0; --i) {
      float a = 0.0f;
      for (int k2 = i + 1; k2 <= t; ++k2) a += R[i][k2] * xc[k2];
      xc[i] = -a / R[i][i];
    }
  }
  __syncthreads();
  for (int i = 0; i < 64; ++i)
    Rinv[i * 64 + t] = (t < KD && i <= t) ? xc[i] : 0.0f;
}

// ---------------------------------------------------------------------------
// 60x60 Jacobi eigensolve of Mm, sort desc, P = ub50 @ (ub50^T @ Zw).
// ---------------------------------------------------------------------------
__global__ void jacobi_pk_kernel(const float* __restrict__ Mm, const float* __restrict__ Zw,
                                 float* __restrict__ P) {
  __shared__ float a[64][64], v[64][64], t1[64][64];
  __shared__ int best;
  const int t = threadIdx.x;  // 64 threads
  for (int i = 0; i < 64; ++i) { a[i][t] = Mm[i * 64 + t]; v[i][t] = (i == t) ? 1.0f : 0.0f; }
  __syncthreads();
  for (int sweep = 0; sweep < 8; ++sweep)
    for (int p = 0; p < KD - 1; ++p)
      for (int q = p + 1; q < KD; ++q) {
        float apq = a[p][q];
        if (fabsf(apq) > 1e-12f) {       // uniform branch (shared value)
          float app = a[p][p], aqq = a[q][q];
          float tau = (aqq - app) / (2.0f * apq);
          float tt = ((tau >= 0.0f) ? 1.0f : -1.0f) / (fabsf(tau) + sqrtf(1.0f + tau * tau));
          float c = 1.0f / sqrtf(1.0f + tt * tt), s = tt * c;
          if (t < KD) {
            float aip = a[t][p], aiq = a[t][q];
            a[t][p] = c * aip - s * aiq;  a[t][q] = s * aip + c * aiq;
          }
          __syncthreads();
          if (t < KD) {
            float api = a[p][t], aqi = a[q][t];
            a[p][t] = c * api - s * aqi;  a[q][t] = s * api + c * aqi;
            float vip = v[t][p], viq = v[t][q];
            v[t][p] = c * vip - s * viq;  v[t][q] = s * vip + c * viq;
          }
          __syncthreads();
        }
      }
  __syncthreads();
  for (int i = 0; i < KD; ++i) {          // selection sort, descending
    if (t == 0) {
      int b = i; float bv = a[i][i];
      for (int j = i + 1; j < KD; ++j) if (a[j][j] > bv) { bv = a[j][j]; b = j; }
      best = b;
    }
    __syncthreads();
    int b = best;
    if (b != i) {
      if (t == 0) { float tmp = a[i][i]; a[i][i] = a[b][b]; a[b][b] = tmp; }
      float tmp = v[t][i]; v[t][i] = v[t][b]; v[t][b] = tmp;
    }
    __syncthreads();
  }
  for (int r = 0; r < RANK; ++r) {        // t1 = ub50^T Zw
    float s2 = 0.0f;
    for (int i2 = 0; i2 < KD; ++i2) s2 += v[i2][r] * Zw[i2 * 64 + t];
    t1[r][t] = s2;
  }
  __syncthreads();
  for (int i2 = 0; i2 < 64; ++i2) {       // P = ub50 t1
    float s2 = 0.0f;
    if (i2 < KD) for (int r = 0; r < RANK; ++r) s2 += v[i2][r] * t1[r][t];
    P[i2 * 64 + t] = s2;
  }
}

// ---------------------------------------------------------------------------
// Fused MLP + pooling: hid = lrelu(term@W1+b1); out[g] += hid@W2.
// term tile TDM-staged; hid kept in LDS (pitch 33, conflict-free).
// ---------------------------------------------------------------------------
__global__ void final_kernel(const float* __restrict__ term, const float* __restrict__ W1,
                             const float* __restrict__ b1, const float* __restrict__ W2,
                             const int* __restrict__ gid, float* __restrict__ out, int n) {
  __shared__ float sT[64 * TPITCH];
  __shared__ float s_hid[64][33];
  __shared__ int s_g[64];
  const int tid = threadIdx.x;
  const int wave = tid >> 5, lane = tid & 31;
  const int half = lane >> 4, lid = lane & 15;
  const int ka = half ? 2 : 0;
  const int base = blockIdx.x * 64;
  if (tid < 64) {
    int r = base + tid;
    s_g[tid] = (r < n) ? gid[r] : -1;
  }
#if USE_TDM
  if (wave == 0) {
    tdm_load_tile_f32(lds_off(&sT[0]), term + (size_t)base * 64, n - base, 64);
    __builtin_amdgcn_s_wait_tensorcnt(0);
  }
  __syncthreads();
#else
  coop_load_tile(sT, term, base, 64, n, tid, 256);
  __syncthreads();
#endif
  {  // stage 1: 8 waves -> 4 row-tiles x 2 col-tiles of hid[64,32]
    const int mt = wave & 3, ct = wave >> 2;
    const float* As = &sT[(mt * 16 + lid) * TPITCH];
    v8f acc = {};
    for (int kk = 0; kk < 16; ++kk) {
      const int k0 = kk * 4;
      v2f a; a.x = As[k0 + ka]; a.y = As[k0 + ka + 1];
      v2f b; b.x = W1[(k0 + ka) * HIDN + ct * 16 + lid];
             b.y = W1[(k0 + ka + 1) * HIDN + ct * 16 + lid];
      acc = wmma4(a, b, acc);
    }
    float bias = b1[ct * 16 + lid];
#pragma unroll
    for (int v = 0; v < 8; ++v) {
      int rr = mt * 16 + v + 8 * half;
      float h = acc[v] + bias;
      s_hid[rr][ct * 16 + lid] = (h > 0.0f) ? h : 0.01f * h;   // leaky_relu(0.01)
    }
  }
  __syncthreads();
  // stage 2: 4 row-tiles x 44 col-tiles (696 cols) over 8 waves
  for (int tile = wave; tile < 4 * 44; tile += 8) {
    const int mt = tile & 3, ct = tile >> 2;
    v8f acc = {};
#pragma unroll
    for (int kk = 0; kk < 8; ++kk) {
      const int k0 = kk * 4;
      v2f a; a.x = s_hid[mt * 16 + lid][k0 + ka];
             a.y = s_hid[mt * 16 + lid][k0 + ka + 1];
      int col = ct * 16 + lid; int colc = (col < NCLS) ? col : (NCLS - 1);
      v2f b; b.x = W2[(k0 + ka) * NCLS + colc];
             b.y = W2[(k0 + ka + 1) * NCLS + colc];
      acc = wmma4(a, b, acc);
    }
    int col = ct * 16 + lid;
    if (col < NCLS) {
#pragma unroll
      for (int v = 0; v < 8; ++v) {
        int g = s_g[mt * 16 + v + 8 * half];
        if (g >= 0) atomicAdd(&out[(size_t)g * NCLS + col], acc[v]);
      }
    }
  }
}

// ---------------------------------------------------------------------------
extern "C" void kernel_launch(void* const* d_in, const int* in_sizes, int n_in,
                              void* d_out, int out_size, void* d_ws, size_t ws_size,
                              hipStream_t stream) {
  const float* x  = (const float*)d_in[0];
  const int* rows = (const int*)d_in[1];
  const int* cols = (const int*)d_in[2];
  const int* gid  = (const int*)d_in[3];
  const float* w  = (const float*)d_in[5];
  const float* W1 = (const float*)d_in[6];
  const float* b1 = (const float*)d_in[7];
  const float* W2 = (const float*)d_in[8];
  const int N  = in_sizes[0] / 64;
  const int E  = in_sizes[1];
  const int FD = in_sizes[5] / 64;  // 21*64 = 1344

  float* p = (float*)d_ws;
  float* deg  = p; p += N;
  float* cs   = p; p += (size_t)4 * N;
  float* Bm   = p; p += (size_t)5 * N * 64;
  float* Q    = p; p += (size_t)N * 64;
  float* Y    = p; p += (size_t)N * 64;
  float* term = p; p += (size_t)N * 64;
  float* Z    = p; p += (size_t)FD * 64;
  float* Z2   = p; p += (size_t)FD * 64;
  float* V    = p; p += (size_t)FD * 64;
  float* G    = p; p += 4096;
  float* Rinv = p; p += 4096;
  float* Mm   = p; p += 4096;
  float* Zw   = p; p += 4096;
  float* P    = p; p += 4096;

  const dim3 b256(256);
  const int gr128_N  = (N + 127) / 128;
  const int gr128_FD = (FD + 127) / 128;
  const int chN  = (N + 2047) / 2048;
  const int chFD = (FD + 2047) / 2048;

  // --- graph propagation ---
  hipMemsetAsync(deg, 0, (size_t)N * 4, stream);
  deg_kernel<<<(E + 255) / 256, b256, 0, stream>>>(rows, deg, E);
  scales_kernel<<<(N + 255) / 256, b256, 0, stream>>>(deg, cs, N);
  hipMemsetAsync(Bm, 0, (size_t)5 * N * 64 * 4, stream);
  const int spg = (int)(((long)E * 64 + 255) / 256);
  spmm_kernel<<<spg, b256, 0, stream>>>(x,  cs,         rows, cols, Bm,                      E);
  spmm_kernel<<<spg, b256, 0, stream>>>(Bm, cs,         rows, cols, Bm + 1 * (size_t)N * 64, E);
  spmm_kernel<<<spg, b256, 0, stream>>>(Bm, cs + N,     rows, cols, Bm + 2 * (size_t)N * 64, E);
  spmm_kernel<<<spg, b256, 0, stream>>>(Bm, cs + 2 * (size_t)N, rows, cols, Bm + 3 * (size_t)N * 64, E);
  spmm_kernel<<<spg, b256, 0, stream>>>(Bm, cs + 3 * (size_t)N, rows, cols, Bm + 4 * (size_t)N * 64, E);

  // --- randomized SVD ---
  omega_kernel<<<(FD * 64 + 255) / 256, b256, 0, stream>>>(V, FD);

#define CHOLQR(src, dst, n, chunks, gr)                                        \
  do {                                                                         \
    hipMemsetAsync(G, 0, 4096 * 4, stream);                                    \
    atb_kernel<<<(chunks), b256, 0, stream>>>((src), (src), G, (n));           \
    cholinv_kernel<<<1, 64, 0, stream>>>(G, Rinv);                             \
    gemm_n64_kernel<<<(gr), b256, 0, stream>>>((src), Rinv, (dst), (n));       \
  } while (0)

  h_times_V_kernel<<<gr128_N, b256, 0, stream>>>(x, Bm, cs, V, Y, N);   // Y = h omega
  CHOLQR(Y, Q, N, chN, gr128_N);
  for (int it = 0; it < NPOW; ++it) {
    hipMemsetAsync(Z, 0, (size_t)FD * 64 * 4, stream);
    hT_times_Q_kernel<<<dim3(chN, 21), b256, 0, stream>>>(x, Bm, cs, Q, Z, N);
    CHOLQR(Z, Z2, FD, chFD, gr128_FD);
    h_times_V_kernel<<<gr128_N, b256, 0, stream>>>(x, Bm, cs, Z2, Y, N);
    CHOLQR(Y, Q, N, chN, gr128_N);
  }
  hipMemsetAsync(Z, 0, (size_t)FD * 64 * 4, stream);
  hT_times_Q_kernel<<<dim3(chN, 21), b256, 0, stream>>>(x, Bm, cs, Q, Z, N);
  hipMemsetAsync(Mm, 0, 4096 * 4, stream);
  atb_kernel<<<chFD, b256, 0, stream>>>(Z, Z, Mm, FD);
  hipMemsetAsync(Zw, 0, 4096 * 4, stream);
  atb_kernel<<<chFD, b256, 0, stream>>>(Z, w, Zw, FD);
  jacobi_pk_kernel<<<1, 64, 0, stream>>>(Mm, Zw, P);
  gemm_n64_kernel<<<gr128_N, b256, 0, stream>>>(Q, P, term, N);

  // --- MLP + pooling ---
  hipMemsetAsync(d_out, 0, (size_t)out_size * 4, stream);
  final_kernel<<<(N + 63) / 64, b256, 0, stream>>>(term, W1, b1, W2, gid, (float*)d_out, N);
#undef CHOLQR
}